// LennardJones_45406394253554
// MI455X (gfx1250) — compile-verified
//
#include <hip/hip_runtime.h>
#include <stdint.h>

#define BLOCK 256
#define LJ_CUTOFF2 100.0f   // 10^2
#define LJ_ONSET2   36.0f   // 6^2

// Issue async DMA of one tile of R_ij (BLOCK pairs * 12 bytes) into LDS.
// Each lane moves its own 12 bytes (b64 + b32); INST_OFFSET is added to both
// the global and the LDS address, so one (ldsaddr, voff) pair covers both.
__device__ __forceinline__ void async_load_tile(const float* __restrict__ R,
                                                int pair0, int tid, int npairs,
                                                float* lds_slot) {
    int p = pair0 + tid;
    if (p >= npairs) p = npairs - 1;            // clamp: harmless duplicate DMA
    uint32_t voff    = (uint32_t)p * 12u;       // byte offset into R_ij stream
    uint32_t ldsaddr = (uint32_t)(uintptr_t)lds_slot; // low 32 bits = LDS offset
    uint64_t base    = (uint64_t)(uintptr_t)R;
    asm volatile("global_load_async_to_lds_b64 %0, %1, %2\n\t"
                 "global_load_async_to_lds_b32 %0, %1, %2 offset:8"
                 :: "v"(ldsaddr), "v"(voff), "s"(base)
                 : "memory");
}

__global__ void __launch_bounds__(BLOCK)
lj_pair_kernel(const float* __restrict__ R,
               const int* __restrict__ idx,
               const uint8_t* __restrict__ pmask,
               const float* __restrict__ sigmaP,
               const float* __restrict__ epsP,
               float* __restrict__ acc,
               int npairs, int ntiles) {
    __shared__ float sR[2][BLOCK * 3];          // 6 KB double buffer

    const int tid    = threadIdx.x;
    const int stride = gridDim.x;
    int t = blockIdx.x;
    if (t >= ntiles) return;

    const float sigma = sigmaP[0];
    const float eps2  = 2.0f * epsP[0];
    const float s2  = sigma * sigma;
    const float s6  = s2 * s2 * s2;
    const float s12 = s6 * s6;
    const float inv_denom = 1.0f /
        ((LJ_CUTOFF2 - LJ_ONSET2) * (LJ_CUTOFF2 - LJ_ONSET2) * (LJ_CUTOFF2 - LJ_ONSET2));

    // Prime the pipeline: DMA tile t into buffer 0.
    async_load_tile(R, t * BLOCK, tid, npairs, &sR[0][tid * 3]);

    int buf = 0;
    for (; t < ntiles; t += stride, buf ^= 1) {
        const int p = t * BLOCK + tid;
        const bool live = p < npairs;
        const int pc = live ? p : npairs - 1;

        // Issue the scalar-stream loads BEFORE waiting on the LDS DMA so their
        // HBM latency overlaps the async wait. Non-temporal: read-once streams
        // must not evict the L2-resident accumulator (400 KB in 192 MB L2).
        const int     node = __builtin_nontemporal_load(&idx[pc]);
        const uint8_t mk   = __builtin_nontemporal_load(&pmask[pc]);

        const int nt = t + stride;
        if (nt < ntiles) {
            // DMA next tile into the other buffer while we compute this one.
            async_load_tile(R, nt * BLOCK, tid, npairs, &sR[buf ^ 1][tid * 3]);
            asm volatile("s_wait_asynccnt 0x2" ::: "memory"); // current tile landed
        } else {
            asm volatile("s_wait_asynccnt 0x0" ::: "memory");
        }

        if (live) {
            const float x = sR[buf][tid * 3 + 0];
            const float y = sR[buf][tid * 3 + 1];
            const float z = sR[buf][tid * 3 + 2];
            const float r2 = x * x + y * y + z * z;

            float contrib = 0.0f;
            if (mk && r2 > 0.0f && r2 < LJ_CUTOFF2) {
                const float inv_r2  = 1.0f / r2;
                const float inv_r6  = inv_r2 * inv_r2 * inv_r2;
                const float inv_r12 = inv_r6 * inv_r6;
                const float pe = eps2 * (s12 * inv_r12 - s6 * inv_r6);
                float sw = 1.0f;
                if (r2 >= LJ_ONSET2) {
                    const float c = LJ_CUTOFF2 - r2;
                    sw = c * c * (LJ_CUTOFF2 + 2.0f * r2 - 3.0f * LJ_ONSET2) * inv_denom;
                }
                contrib = sw * pe;
            }
            if (contrib != 0.0f)
                atomicAdd(&acc[node], contrib);  // global_atomic_add_f32 (no-return)
        }
    }
}

__global__ void zero_kernel(float* __restrict__ acc, int n) {
    int k = blockIdx.x * blockDim.x + threadIdx.x;
    if (k < n) acc[k] = 0.0f;
}

__global__ void finalize_kernel(float* __restrict__ out,
                                const uint8_t* __restrict__ nmask, int n) {
    int k = blockIdx.x * blockDim.x + threadIdx.x;
    if (k < n) out[k] = nmask[k] ? out[k] : 0.0f;
}

extern "C" void kernel_launch(void* const* d_in, const int* in_sizes, int n_in,
                              void* d_out, int out_size, void* d_ws, size_t ws_size,
                              hipStream_t stream) {
    const float*   R     = (const float*)d_in[0];
    const int*     idx   = (const int*)d_in[1];
    // d_in[2] (j) and d_in[3] (Z_i values) are unused by the reference math.
    const uint8_t* pmask = (const uint8_t*)d_in[4];
    const uint8_t* nmask = (const uint8_t*)d_in[5];
    const float*   sigma = (const float*)d_in[6];
    const float*   eps   = (const float*)d_in[7];
    float*         out   = (float*)d_out;

    const int npairs = in_sizes[1];   // i has one entry per pair
    const int nnodes = in_sizes[3];   // Z_i has one entry per node

    const int ntiles = (npairs + BLOCK - 1) / BLOCK;
    int grid = ntiles < 4096 ? ntiles : 4096;   // persistent blocks, grid-stride tiles

    zero_kernel<<<(nnodes + 255) / 256, 256, 0, stream>>>(out, nnodes);
    lj_pair_kernel<<<grid, BLOCK, 0, stream>>>(R, idx, pmask, sigma, eps,
                                               out, npairs, ntiles);
    finalize_kernel<<<(nnodes + 255) / 256, 256, 0, stream>>>(out, nmask, nnodes);
}